// FullPrecisionAttentionBlock_87239375716624
// MI455X (gfx1250) — compile-verified
//
#include <hip/hip_runtime.h>
#include <math.h>

#define N_SEQ 1024
#define DE 64
#define DK 16
#define EPSV 1e-5f

typedef __attribute__((ext_vector_type(2))) float v2f;
typedef __attribute__((ext_vector_type(8))) float v8f;

// D = A(16x4,f32) * B(4x16,f32) + C  -- CDNA5 V_WMMA_F32_16X16X4_F32
// 8-arg form: (neg_a, A, neg_b, B, c_mod, C, reuse_a, reuse_b); neg must be 0 for f32.
__device__ __forceinline__ v8f wmma4(v2f a, v2f b, v8f c) {
    return __builtin_amdgcn_wmma_f32_16x16x4_f32(false, a, false, b, (short)0, c, false, false);
}

// ---------------------------------------------------------------------------
// Prep: per-d constants (lam_r, lam_i, Omega, Gamma, coefs, one-step rotation)
// ---------------------------------------------------------------------------
__global__ __launch_bounds__(32)
void prep_kernel(const float* __restrict__ tm, const float* __restrict__ lambda1,
                 const float* __restrict__ lOm, const float* __restrict__ lGa,
                 const float* __restrict__ noise, const float* __restrict__ tau,
                 const float* __restrict__ nu, const float* __restrict__ etap,
                 float* __restrict__ C, float* __restrict__ lamh) {
    int d = threadIdx.x;
    if (d < DK) {
        int k = d & 7;
        float l0 = lambda1[k];
        float l1 = lambda1[8 + k];
        float lr = -fabsf(l0);
        float li = (d < 8) ? l1 : -l1;
        C[0 + d]  = lr;
        C[16 + d] = li;
        float om = fabsf(lOm[d]);
        float ga = fabsf(lGa[d]) + EPSV;
        C[32 + d] = om;
        C[48 + d] = ga;
        C[64 + d] = om / (-2.0f * lr + EPSV);
        float eta = 1.0f / (1.0f + __expf(-etap[d]));       // sigmoid(eta_param)
        C[80 + d] = 1.0f - eta;                             // (1 - eta)
        C[96 + d] = 1.0f / (1.0f + __expf(-eta));           // sigmoid(eta)
        float invT = 1.0f / (tm[N_SEQ - 1] - tm[0]);
        float dt1 = (tm[1] - tm[0]) * invT;
        float mm = __expf(lr * dt1);
        C[112 + d] = mm * __cosf(li * dt1);                 // mr
        C[128 + d] = mm * __sinf(li * dt1);                 // mi
        lamh[d] = lr;                                       // lambda_h[0][d]
        lamh[16 + d] = li;                                  // lambda_h[1][d]
    }
    if (d == 0) {
        C[144] = fabsf(nu[0]);
        C[145] = fabsf(tau[0]);
        C[146] = noise[0];
        C[147] = 1.0f / (tm[N_SEQ - 1] - tm[0]);
    }
}

// ---------------------------------------------------------------------------
// Complex projection via WMMA: Y = clinear(Z, W, b);  Z:(2,N,DE) W:(2,DK,DE)
// out layout: [c][i][d]  (c stride N*DK)
// One wave (32 threads) per 16-row tile; 64 chained v_wmma per block.
// ---------------------------------------------------------------------------
__global__ __launch_bounds__(32)
void proj_kernel(const float* __restrict__ Z, const float* __restrict__ W,
                 const float* __restrict__ bias, float* __restrict__ outw) {
    const int lane = threadIdx.x;
    const int m = lane & 15;        // row within tile (A) / col (B,D)
    const int half = lane >> 4;     // K-pair selector
    const int row0 = blockIdx.x * 16;
    const float* Zr = Z;
    const float* Zi = Z + N_SEQ * DE;
    const float* W0 = W;            // (DK, DE)
    const float* W1 = W + DK * DE;
    v8f accR = {0, 0, 0, 0, 0, 0, 0, 0};
    v8f accI = {0, 0, 0, 0, 0, 0, 0, 0};
    for (int k0 = 0; k0 < DE; k0 += 4) {
        int ka = k0 + 2 * half;
        v2f axr, axi, axin, b0, b1;
        axr[0] = Zr[(row0 + m) * DE + ka];     axr[1] = Zr[(row0 + m) * DE + ka + 1];
        axi[0] = Zi[(row0 + m) * DE + ka];     axi[1] = Zi[(row0 + m) * DE + ka + 1];
        axin[0] = -axi[0];                     axin[1] = -axi[1];
        // B[k][n] = W[n][k]  -> lane holds n=m, K = ka..ka+1
        b0[0] = W0[m * DE + ka];               b0[1] = W0[m * DE + ka + 1];
        b1[0] = W1[m * DE + ka];               b1[1] = W1[m * DE + ka + 1];
        accR = wmma4(axr,  b0, accR);   // + xr @ W0^T
        accR = wmma4(axin, b1, accR);   // - xi @ W1^T
        accI = wmma4(axr,  b1, accI);   // + xr @ W1^T
        accI = wmma4(axi,  b0, accI);   // + xi @ W0^T
    }
    float br = bias[m], bi = bias[DK + m];
    for (int r = 0; r < 8; ++r) {
        int mr = r + 8 * half;          // C/D layout: VGPR r, lane-half selects M
        outw[(row0 + mr) * DK + m] = accR[r] + br;
        outw[N_SEQ * DK + (row0 + mr) * DK + m] = accI[r] + bi;
    }
}

// ---------------------------------------------------------------------------
// Per-row kernel: big (j,d) map, causal softmax over j, est_v reduction.
// Block = 256 threads (8 waves): d = tid%16 (16-lane half), j-slot = tid/16.
// Dynamic LDS: 64KB score row s[1024][16] (CDNA5 allows up to 320KB/WG).
// ---------------------------------------------------------------------------
__global__ __launch_bounds__(256)
void row_kernel(const float* __restrict__ tm, const float* __restrict__ C,
                const float* __restrict__ Qw, const float* __restrict__ Kw,
                const float* __restrict__ Vw, float* __restrict__ Qij,
                float* __restrict__ Zhat, float* __restrict__ estl,
                float* __restrict__ Pws) {
    extern __shared__ float s[];                 // [1024][16]
    __shared__ float red[16][16];
    __shared__ float g0[16], g1[16];
    const int i = blockIdx.x;
    const int tid = threadIdx.x;
    const int d = tid & 15;
    const int slot = tid >> 4;
    const float lamr = C[d],   lami = C[16 + d];
    const float ga = C[48 + d], c2 = C[64 + d];
    const float absNu = C[144], absTau = C[145], noise = C[146], invT = C[147];
    const float ti = tm[i];
    const float qr = Qw[i * DK + d], qi = Qw[N_SEQ * DK + i * DK + d];
    const size_t NN = (size_t)N_SEQ * N_SEQ * DK;

    // ---- pass 1: scores + Z_hat ----
    float lmax = -INFINITY;
    for (int it = 0; it < 64; ++it) {
        int jj = slot + (it << 4);
        bool valid = (jj <= i);
        float zr = 0.f, zi = 0.f, mah = 0.f, vij = 0.f;
        if (valid) {
            float dt = (ti - tm[jj]) * invT;     // >= 0 in causal region
            float mag = __expf(lamr * dt);
            float ph = lami * dt;
            float cr = __cosf(ph), sn = __sinf(ph);
            float er = mag * cr, ei = mag * sn;
            float vr = Vw[jj * DK + d], vi2 = Vw[N_SEQ * DK + jj * DK + d];
            float kr = Kw[jj * DK + d], ki  = Kw[N_SEQ * DK + jj * DK + d];
            zr = er * vr - ei * vi2;
            zi = er * vi2 + ei * vr;
            float rr = er * kr - ei * ki - qr;
            float ri = er * ki + ei * kr - qi;
            float a = mag * mag;                 // exp(2*lam_r*|dt|)
            vij = ga * a + c2 * (1.0f - a);
            mah = (rr * rr + ri * ri) / (vij + ga);
        }
        // sum mahal over the 16 d-lanes (stays within wave32 half)
        float dist = mah;
        dist += __shfl_xor(dist, 1);
        dist += __shfl_xor(dist, 2);
        dist += __shfl_xor(dist, 4);
        dist += __shfl_xor(dist, 8);
        float sc = -INFINITY;
        if (valid) {
            float base = vij * (noise + absNu * dist + EPSV);
            sc = -absTau * __logf(base);
        }
        s[jj * DK + d] = sc;
        lmax = fmaxf(lmax, sc);
        size_t zo = ((size_t)i * N_SEQ + jj) * DK + d;
        Zhat[zo] = zr;                           // masked: 0 above diagonal
        Zhat[NN + zo] = zi;
    }
    // ---- block max per d ----
    red[slot][d] = lmax;
    __syncthreads();
    if (tid < 16) {
        float v = red[0][tid];
        for (int ss = 1; ss < 16; ++ss) v = fmaxf(v, red[ss][tid]);
        g0[tid] = v;
    }
    __syncthreads();
    float gm = g0[d];
    // ---- pass 2a: exp + sum ----
    float lsum = 0.f;
    for (int it = 0; it < 64; ++it) {
        int jj = slot + (it << 4);
        float e = __expf(s[jj * DK + d] - gm);   // exp(-inf)=0 above diagonal
        s[jj * DK + d] = e;
        lsum += e;
    }
    red[slot][d] = lsum;
    __syncthreads();
    if (tid < 16) {
        float v = 0.f;
        for (int ss = 0; ss < 16; ++ss) v += red[ss][tid];
        g1[tid] = (v > 0.f) ? 1.0f / v : 0.f;
    }
    __syncthreads();
    float rden = g1[d];
    // ---- pass 2b: write Q_ij, accumulate est_v (re-read L2-hot Z_hat) ----
    float accr = 0.f, acci = 0.f;
    for (int it = 0; it < 64; ++it) {
        int jj = slot + (it << 4);
        float q = s[jj * DK + d] * rden;
        size_t qo = ((size_t)i * N_SEQ + jj) * DK + d;
        Qij[qo] = q;
        if (jj <= i) {
            float zr = Zhat[qo], zi = Zhat[NN + qo];
            accr += q * zr;
            acci += q * zi;
        }
    }
    // ---- reduce est_v per d ----
    __syncthreads();
    red[slot][d] = accr;
    __syncthreads();
    if (tid < 16) {
        float v = 0.f;
        for (int ss = 0; ss < 16; ++ss) v += red[ss][tid];
        g0[tid] = v;
    }
    __syncthreads();
    red[slot][d] = acci;
    __syncthreads();
    if (tid < 16) {
        float v = 0.f;
        for (int ss = 0; ss < 16; ++ss) v += red[ss][tid];
        g1[tid] = v;
    }
    __syncthreads();
    if (tid < 16) {
        int dd = tid;
        float cA = C[80 + dd], cB = C[96 + dd];
        float vr = Vw[i * DK + dd], vi2 = Vw[N_SEQ * DK + i * DK + dd];
        float er = cA * vr + cB * g0[dd];        // est_latent real
        float ei = cA * vi2 + cB * g1[dd];       // est_latent imag
        estl[i * DK + dd] = er;
        estl[N_SEQ * DK + i * DK + dd] = ei;
        float mr = C[112 + dd], mi = C[128 + dd];
        Pws[i * DK + dd] = mr * er - mi * ei;    // pred_p real
        Pws[N_SEQ * DK + i * DK + dd] = mr * ei + mi * er;
    }
}

// ---------------------------------------------------------------------------
// Final complex projection via WMMA: out = clinear(pred_p, Wp, bp)
// P:(2,N,DK)  Wp:(2,DE,DK)  out:(2,N,DE).  grid=(row tiles, col tiles)
// ---------------------------------------------------------------------------
__global__ __launch_bounds__(32)
void outproj_kernel(const float* __restrict__ P, const float* __restrict__ Wp,
                    const float* __restrict__ bp, float* __restrict__ out) {
    const int lane = threadIdx.x;
    const int m = lane & 15;
    const int half = lane >> 4;
    const int row0 = blockIdx.x * 16;
    const int e0 = blockIdx.y * 16;
    const float* Pr = P;
    const float* Pi = P + N_SEQ * DK;
    const float* W0 = Wp;                // (DE, DK)
    const float* W1 = Wp + DE * DK;
    v8f accR = {0, 0, 0, 0, 0, 0, 0, 0};
    v8f accI = {0, 0, 0, 0, 0, 0, 0, 0};
    for (int k0 = 0; k0 < DK; k0 += 4) {
        int ka = k0 + 2 * half;
        v2f axr, axi, axin, b0, b1;
        axr[0] = Pr[(row0 + m) * DK + ka];  axr[1] = Pr[(row0 + m) * DK + ka + 1];
        axi[0] = Pi[(row0 + m) * DK + ka];  axi[1] = Pi[(row0 + m) * DK + ka + 1];
        axin[0] = -axi[0];                  axin[1] = -axi[1];
        b0[0] = W0[(e0 + m) * DK + ka];     b0[1] = W0[(e0 + m) * DK + ka + 1];
        b1[0] = W1[(e0 + m) * DK + ka];     b1[1] = W1[(e0 + m) * DK + ka + 1];
        accR = wmma4(axr,  b0, accR);
        accR = wmma4(axin, b1, accR);
        accI = wmma4(axr,  b1, accI);
        accI = wmma4(axi,  b0, accI);
    }
    float br = bp[e0 + m], bi = bp[DE + e0 + m];
    for (int r = 0; r < 8; ++r) {
        int mr = r + 8 * half;
        out[(row0 + mr) * DE + e0 + m] = accR[r] + br;
        out[N_SEQ * DE + (row0 + mr) * DE + e0 + m] = accI[r] + bi;
    }
}

// ---------------------------------------------------------------------------
extern "C" void kernel_launch(void* const* d_in, const int* in_sizes, int n_in,
                              void* d_out, int out_size, void* d_ws, size_t ws_size,
                              hipStream_t stream) {
    (void)in_sizes; (void)n_in; (void)out_size; (void)ws_size;
    const float* Zq = (const float*)d_in[0];
    const float* Zk = (const float*)d_in[1];
    const float* Zv = (const float*)d_in[2];
    const float* tm = (const float*)d_in[3];
    const float* Wq = (const float*)d_in[4];
    const float* bq = (const float*)d_in[5];
    const float* Wk = (const float*)d_in[6];
    const float* bk = (const float*)d_in[7];
    const float* Wv = (const float*)d_in[8];
    const float* bv = (const float*)d_in[9];
    const float* Wp = (const float*)d_in[10];
    const float* bp = (const float*)d_in[11];
    const float* lambda1 = (const float*)d_in[12];
    const float* lOm = (const float*)d_in[13];
    const float* lGa = (const float*)d_in[14];
    const float* noise = (const float*)d_in[15];
    const float* tau = (const float*)d_in[16];
    const float* nu = (const float*)d_in[17];
    const float* etap = (const float*)d_in[18];

    float* outf = (float*)d_out;
    float* estl = outf;                      // (1,2,1024,16)    32768
    float* outp = outf + 32768;              // (1,2,1024,64)   131072
    float* Qij  = outf + 163840;             // (1,1024,1024,16) 16777216
    float* Zhat = outf + 16941056;           // (1,2,1024,1024,16) 33554432
    float* lamh = outf + 50495488;           // (2,16,1)         32

    float* ws  = (float*)d_ws;
    float* Qw  = ws;                         // (2,N,DK)
    float* Kw  = ws + 32768;
    float* Vw  = ws + 65536;
    float* Pws = ws + 98304;
    float* C   = ws + 131072;                // per-d constants + scalars

    prep_kernel<<<1, 32, 0, stream>>>(tm, lambda1, lOm, lGa, noise, tau, nu, etap, C, lamh);
    proj_kernel<<<64, 32, 0, stream>>>(Zq, Wq, bq, Qw);
    proj_kernel<<<64, 32, 0, stream>>>(Zk, Wk, bk, Kw);
    proj_kernel<<<64, 32, 0, stream>>>(Zv, Wv, bv, Vw);
    row_kernel<<<1024, 256, 1024 * DK * sizeof(float), stream>>>(
        tm, C, Qw, Kw, Vw, Qij, Zhat, estl, Pws);
    outproj_kernel<<<dim3(64, 4), 32, 0, stream>>>(Pws, Wp, bp, outp);
}